// evalYolov2_60748017435091
// MI455X (gfx1250) — compile-verified
//
#include <hip/hip_runtime.h>
#include <stdint.h>

// ---------------- problem constants ----------------
#define NUMA    5
#define CLASSES 80
#define ALEN    85          // CLASSES + 4 + 1
#define HH      38
#define WW      38
#define NBOX    (NUMA*HH*WW)   // 7220
#define N2      7232           // NBOX padded to multiple of 16 (and 32)
#define NT      452            // ceil(NBOX/16): 16-wide column tiles
#define NWORDS  226            // ceil(NBOX/32): 32-bit suppression words
#define NIMG    4
#define SORTN   8192
#define NMS_T   0.45f

#if __has_builtin(__builtin_amdgcn_global_load_async_to_lds_b32) && \
    __has_builtin(__builtin_amdgcn_s_wait_asynccnt)
#define USE_ASYNC 1
#else
#define USE_ASYNC 0
#endif

// clang (HIP mode) prints AS1 as `__device__` and AS3 as `__shared__`; the
// async-to-LDS builtin wants (AS1 int*, AS3 int*, imm, imm).
#define AS1CAST(p) ((__attribute__((address_space(1))) int*)(p))
#define AS3CAST(p) ((__attribute__((address_space(3))) int*)(p))

__constant__ float ANCH[10] = {0.57273f, 0.677385f, 1.87446f, 2.06253f,
                               3.33843f, 5.47434f, 7.88282f, 3.52778f,
                               9.77052f, 9.16828f};

typedef __attribute__((ext_vector_type(2))) float v2f;
typedef __attribute__((ext_vector_type(8))) float v8f;

// ---------------- 1) decode: raw pred -> (obj, box, prob, cls), planar per image ----------------
__global__ void decode_k(const float* __restrict__ pred,
                         float* __restrict__ objU, float* __restrict__ boxU,
                         float* __restrict__ probU, float* __restrict__ clsU) {
  int t = blockIdx.x * blockDim.x + threadIdx.x;
  if (t >= NIMG * NBOX) return;
  int img = t / NBOX, q = t % NBOX;
  int a = q / (HH * WW), r = q % (HH * WW);
  int gj = r / WW, gi = r % WW;
  const int S = HH * WW;
  const float* base = pred + (size_t)img * (NUMA * ALEN * S) + (size_t)(a * ALEN) * S + gj * WW + gi;

  float tx = base[0 * S], ty = base[1 * S], tw = base[2 * S], th = base[3 * S], to = base[4 * S];
  float bx = (tx + (float)gi) / (float)WW;
  float by = (ty + (float)gj) / (float)HH;
  float bw = expf(tw) * (ANCH[2 * a]     / (float)WW);
  float bh = expf(th) * (ANCH[2 * a + 1] / (float)HH);

  __builtin_prefetch(base + 5 * S, 0, 3);   // gfx1250: global_prefetch_b8
  float best = base[5 * S];
  int bc = 0;
  for (int c = 1; c < CLASSES; ++c) {
    float v = base[(5 + c) * S];
    if (v > best) { best = v; bc = c; }
  }

  size_t NN = (size_t)NIMG * N2;
  size_t o = (size_t)img * N2 + q;
  objU [o]          = to;
  boxU [0*NN + o]   = bx;
  boxU [1*NN + o]   = by;
  boxU [2*NN + o]   = bw;
  boxU [3*NN + o]   = bh;
  probU[o]          = best;
  clsU [o]          = (float)bc;
}

// ---------------- 2) stable argsort(-obj) via in-LDS bitonic sort of packed 64-bit keys ----------------
// key = (descending-ordered obj bits) << 32 | index  -> ascending sort == stable argsort(-obj)
__global__ void sort_k(const float* __restrict__ objU, uint32_t* __restrict__ order) {
  __shared__ uint64_t sk[SORTN];   // 64 KB of the 320 KB/WGP LDS
  int img = blockIdx.x, tid = threadIdx.x;
  const float* ob = objU + (size_t)img * N2;

  for (int t = tid; t < SORTN; t += blockDim.x) {
    uint64_t key;
    if (t < NBOX) {
      uint32_t u = __float_as_uint(ob[t]);
      u = (u & 0x80000000u) ? ~u : (u | 0x80000000u);   // monotone float->uint (ascending)
      uint32_t kd = ~u;                                  // descending obj
      key = ((uint64_t)kd << 32) | (uint32_t)t;
    } else {
      key = ~0ull;                                       // pad sorts last
    }
    sk[t] = key;
  }
  __syncthreads();

  for (int k = 2; k <= SORTN; k <<= 1) {
    for (int j = k >> 1; j > 0; j >>= 1) {
      for (int p = tid; p < SORTN / 2; p += blockDim.x) {
        int i   = ((p & ~(j - 1)) << 1) | (p & (j - 1));
        int ixj = i | j;
        bool up = ((i & k) == 0);
        uint64_t va = sk[i], vb = sk[ixj];
        if ((va > vb) == up) { sk[i] = vb; sk[ixj] = va; }
      }
      __syncthreads();
    }
  }

  for (int t = tid; t < NBOX; t += blockDim.x)
    order[(size_t)img * N2 + t] = (uint32_t)sk[t];
}

// ---------------- 3) gather sorted arrays + corners + areas ----------------
__global__ void gather_k(const uint32_t* __restrict__ order,
                         const float* __restrict__ objU, const float* __restrict__ boxU,
                         const float* __restrict__ probU, const float* __restrict__ clsU,
                         float* __restrict__ objS, float* __restrict__ boxS,
                         float* __restrict__ probS, float* __restrict__ clsS,
                         float* __restrict__ corn, float* __restrict__ areaS) {
  int t = blockIdx.x * blockDim.x + threadIdx.x;
  if (t >= NIMG * NBOX) return;
  int img = t / NBOX, q = t % NBOX;
  size_t NN = (size_t)NIMG * N2;
  size_t ib = (size_t)img * N2;
  uint32_t s = order[ib + q];

  float bx = boxU[0*NN + ib + s], by = boxU[1*NN + ib + s];
  float bw = boxU[2*NN + ib + s], bh = boxU[3*NN + ib + s];
  objS [ib + q]        = objU [ib + s];
  probS[ib + q]        = probU[ib + s];
  clsS [ib + q]        = clsU [ib + s];
  boxS [0*NN + ib + q] = bx;
  boxS [1*NN + ib + q] = by;
  boxS [2*NN + ib + q] = bw;
  boxS [3*NN + ib + q] = bh;
  corn [0*NN + ib + q] = bx - 0.5f * bw;   // x1
  corn [1*NN + ib + q] = by - 0.5f * bh;   // y1
  corn [2*NN + ib + q] = bx + 0.5f * bw;   // x2
  corn [3*NN + ib + q] = by + 0.5f * bh;   // y2
  areaS[ib + q]        = bw * bh;
}

// ---------------- 4) suppression bitmask, 16x16 tiles, 8 waves/block, one wave32 per tile ----------------
// union_base[i][j] = area_i + area_j computed as a rank-2 WMMA:
//   A (16x4): [area_i, 1, 0, 0]   B (4x16): row0 = 1..1, row1 = area_j
//   D = A x B + 0  ==> D[i][j] = area_i + area_j  (V_WMMA_F32_16X16X4_F32)
__global__ void mask_k(const float* __restrict__ corn, const float* __restrict__ areaS,
                       uint16_t* __restrict__ mask) {
  int wave = threadIdx.x >> 5;            // 0..7
  int lane = threadIdx.x & 31;
  int J = blockIdx.x * 8 + wave;          // column tile (wave-uniform)
  int I = blockIdx.y, img = blockIdx.z;
  if (J >= NT) return;
  int i0 = I * 16, j0 = J * 16;
  uint16_t* mimg = mask + (size_t)img * N2 * NT;

  if (J < I) {                            // lower triangle: zero-fill so the reducer can read full rows
    if (lane < 16) {
      int i = i0 + lane;
      if (i < NBOX) mimg[(size_t)i * NT + J] = 0;
    }
    return;
  }

  size_t NN = (size_t)NIMG * N2;
  size_t ib = (size_t)img * N2;
  const float* c1x = corn + 0*NN + ib;
  const float* c1y = corn + 1*NN + ib;
  const float* c2x = corn + 2*NN + ib;
  const float* c2y = corn + 3*NN + ib;
  const float* ar  = areaS + ib;

  bool lo = lane < 16;
  int  l15 = lane & 15;

  // A-matrix layout (16x4 f32): lanes 0-15 hold M=lane, VGPR0=K0, VGPR1=K1; lanes 16-31 hold K2,K3.
  // B-matrix layout (4x16 f32): lanes 0-15 hold N=lane, VGPR0=K0 row, VGPR1=K1 row; lanes 16-31 K2,K3.
  v2f A, B;
  if (lo) {
    A.x = ar[i0 + l15];  A.y = 1.0f;         // [area_i, 1]
    B.x = 1.0f;          B.y = ar[j0 + l15]; // [1, area_j]
  } else {
    A.x = 0.0f; A.y = 0.0f; B.x = 0.0f; B.y = 0.0f;   // K=2,3 contribute nothing
  }
  v8f C = {0.f, 0.f, 0.f, 0.f, 0.f, 0.f, 0.f, 0.f};
  v8f D = __builtin_amdgcn_wmma_f32_16x16x4_f32(false, A, false, B, (short)0, C, false, false);

  // column j for this lane (N = lane & 15); loads stay in-bounds thanks to N2 padding
  int j = j0 + l15;
  float jx1 = c1x[j], jy1 = c1y[j], jx2 = c2x[j], jy2 = c2y[j];

  unsigned int bals[8];
  #pragma unroll
  for (int v = 0; v < 8; ++v) {
    int irow = i0 + (lo ? v : v + 8);     // D layout: lanes 0-15 -> M=v, lanes 16-31 -> M=v+8
    float ix1 = c1x[irow], iy1 = c1y[irow], ix2 = c2x[irow], iy2 = c2y[irow];
    float L = fmaxf(ix1, jx1), U = fmaxf(iy1, jy1);
    float R = fminf(ix2, jx2), Dn = fminf(iy2, jy2);
    float inter = fmaxf(R - L, 0.f) * fmaxf(Dn - U, 0.f);
    float uni   = D[v] - inter;           // area_i + area_j - inter (from WMMA)
    float iou   = inter / uni;
    bool bit = (irow < NBOX) && (j < NBOX) && (j > irow) && (iou >= NMS_T);
    bals[v] = (unsigned int)__ballot(bit);   // wave32: bits 0-15 = row v, bits 16-31 = row v+8
  }

  if (lane < 8) {
    int i = i0 + lane;
    if (i < NBOX) mimg[(size_t)i * NT + J] = (uint16_t)(bals[lane] & 0xFFFFu);
  } else if (lane < 16) {
    int i = i0 + lane;
    if (i < NBOX) mimg[(size_t)i * NT + J] = (uint16_t)(bals[lane - 8] >> 16);
  }
}

// ---------------- 5) sequential greedy reduce over bitmask + emit outputs ----------------
// The serial loop is latency-bound on the dependent row fetch; on gfx1250 we
// double-buffer the next mask row into LDS with global_load_async_to_lds_b32
// (ASYNCcnt) so the L2 round-trip overlaps the flag/barrier logic. Each wave
// consumes only the words it issued, so s_wait_asynccnt(1) + block barrier
// gives sufficient ordering.
__global__ void nms_out_k(const uint16_t* __restrict__ mask,
                          const float* __restrict__ objS, const float* __restrict__ boxS,
                          const float* __restrict__ probS, const float* __restrict__ clsS,
                          float* __restrict__ out) {
  int img = blockIdx.x, tid = threadIdx.x;    // blockDim = 512
  __shared__ uint32_t rem[NWORDS];
  __shared__ int skip;                        // 0 = active row, 1 = skip, 2 = stop (obj <= 0)
#if USE_ASYNC
  __shared__ uint32_t rowbuf[2][NWORDS];
#endif

  for (int w = tid; w < NWORDS; w += blockDim.x) rem[w] = 0;
  __syncthreads();

  size_t ib = (size_t)img * N2;
  const uint16_t* m = mask + (size_t)img * N2 * NT;

#if USE_ASYNC
  if (tid < NWORDS) {                         // prefetch row 0
    const uint32_t* g0 = (const uint32_t*)m + tid;   // NT*2 bytes/row, 4B aligned
    __builtin_amdgcn_global_load_async_to_lds_b32(AS1CAST((void*)g0),
                                                  AS3CAST(&rowbuf[0][tid]), 0, 0);
  }
#endif

  for (int i = 0; i < NBOX; ++i) {
#if USE_ASYNC
    if (i + 1 < NBOX && tid < NWORDS) {       // kick off row i+1 while row i is consumed
      const uint32_t* gn = (const uint32_t*)(m + (size_t)(i + 1) * NT) + tid;
      __builtin_amdgcn_global_load_async_to_lds_b32(AS1CAST((void*)gn),
                                                    AS3CAST(&rowbuf[(i + 1) & 1][tid]), 0, 0);
    }
    __builtin_amdgcn_s_wait_asynccnt(1);      // row i landed in LDS (per-wave in-order)
#endif
    if (tid == 0) {
      int r = (rem[i >> 5] >> (i & 31)) & 1;
      float o = objS[ib + i];
      skip = (o <= 0.f) ? 2 : r;              // sorted desc: once obj<=0 nothing else is active
    }
    __syncthreads();
    if (skip == 2) break;
    if (skip == 0 && tid < NWORDS) {
#if USE_ASYNC
      rem[tid] |= rowbuf[i & 1][tid];         // mask only holds bits with j > i
#else
      const uint16_t* row = m + (size_t)i * NT;
      rem[tid] |= (uint32_t)row[2 * tid] | ((uint32_t)row[2 * tid + 1] << 16);
#endif
    }
    __syncthreads();
  }
  __syncthreads();

  size_t NN = (size_t)NIMG * N2;
  float* det    = out + (size_t)img * NBOX * 6;
  float* clsOut = out + (size_t)NIMG * NBOX * 6 + (size_t)img * NBOX;
  for (int q = tid; q < NBOX; q += blockDim.x) {
    int r = (rem[q >> 5] >> (q & 31)) & 1;
    det[q * 6 + 0] = r ? 0.f : objS[ib + q];
    det[q * 6 + 1] = boxS[0*NN + ib + q];
    det[q * 6 + 2] = boxS[1*NN + ib + q];
    det[q * 6 + 3] = boxS[2*NN + ib + q];
    det[q * 6 + 4] = boxS[3*NN + ib + q];
    det[q * 6 + 5] = probS[ib + q];
    clsOut[q]      = clsS[ib + q];
  }
}

// ---------------- host launcher ----------------
extern "C" void kernel_launch(void* const* d_in, const int* in_sizes, int n_in,
                              void* d_out, int out_size, void* d_ws, size_t ws_size,
                              hipStream_t stream) {
  const float* pred = (const float*)d_in[0];
  float* out = (float*)d_out;

  // workspace carve-out (~28.5 MB total):
  //  floats: objU[NN] boxU[4NN] probU[NN] clsU[NN] | order u32[NN] |
  //          objS[NN] boxS[4NN] probS[NN] clsS[NN] corn[4NN] areaS[NN]
  //  mask:   uint16[NIMG * N2 * NT]
  size_t NN = (size_t)NIMG * N2;
  float*    objU  = (float*)d_ws;
  float*    boxU  = objU + NN;
  float*    probU = boxU + 4 * NN;
  float*    clsU  = probU + NN;
  uint32_t* order = (uint32_t*)(clsU + NN);
  float*    objS  = (float*)(order + NN);
  float*    boxS  = objS + NN;
  float*    probS = boxS + 4 * NN;
  float*    clsS  = probS + NN;
  float*    corn  = clsS + NN;
  float*    areaS = corn + 4 * NN;
  uint16_t* mask  = (uint16_t*)(areaS + NN);

  int totalCells = NIMG * NBOX;
  decode_k<<<(totalCells + 255) / 256, 256, 0, stream>>>(pred, objU, boxU, probU, clsU);
  sort_k<<<NIMG, 1024, 0, stream>>>(objU, order);
  gather_k<<<(totalCells + 255) / 256, 256, 0, stream>>>(order, objU, boxU, probU, clsU,
                                                         objS, boxS, probS, clsS, corn, areaS);
  mask_k<<<dim3((NT + 7) / 8, NT, NIMG), 256, 0, stream>>>(corn, areaS, mask);
  nms_out_k<<<NIMG, 512, 0, stream>>>(mask, objS, boxS, probS, clsS, out);
}